// PosGCN_84602265796922
// MI455X (gfx1250) — compile-verified
//
#include <hip/hip_runtime.h>

typedef __attribute__((ext_vector_type(2))) float v2f;
typedef __attribute__((ext_vector_type(8))) float v8f;

#define EMBED 128

// ---------------------------------------------------------------------------
// h = x @ W  using V_WMMA_F32_16X16X4_F32.
// Block = 256 threads = 8 waves. Block handles a 16-row strip of x; wave w
// produces the 16x16 tile at columns [16w, 16w+16). K loop: 128 in steps of 4.
//
// A (16x4 f32) layout per ISA: lanes 0-15 hold M=lane with K={0,1} in the two
// VGPRs; lanes 16-31 hold M=lane-16 with K={2,3}.
// B (4x16) mirrors: VGPR0 = row K=kk, VGPR1 = row K=kk+1 at column c0+lane%16.
// C/D (16x16 f32): VGPR v -> row v (lanes 0-15) / row v+8 (lanes 16-31),
// column = c0 + lane%16.
// ---------------------------------------------------------------------------
__global__ __launch_bounds__(256) void gemm_xW_wmma(const float* __restrict__ x,
                                                    const float* __restrict__ W,
                                                    float* __restrict__ h,
                                                    int nrows) {
  __shared__ float xs[16 * EMBED];   // 8 KB x-tile
  const int tid = threadIdx.x;
  const int r0  = blockIdx.x * 16;
  if (r0 + 16 > nrows) return;       // N=100000 is divisible by 16; guard anyway

  // Cooperative coalesced load: tile spans full row width -> 2048 contiguous f32.
  {
    const float4* src = (const float4*)(x + (size_t)r0 * EMBED);
    float4* dst = (float4*)xs;
    dst[tid]       = src[tid];
    dst[tid + 256] = src[tid + 256];
  }
  __syncthreads();

  const int wave  = tid >> 5;
  const int lane  = tid & 31;
  const int c0    = wave * 16;       // this wave's column tile
  const int mlane = lane & 15;
  const int hi    = lane >> 4;       // 0 for lanes 0-15, 1 for lanes 16-31
  const int kk    = hi * 2;

  v8f acc = {};
  #pragma unroll
  for (int k = 0; k < EMBED; k += 4) {
    // A fragment from LDS (b64 load, 8B aligned: k%4==0, kk in {0,2})
    v2f a = *(const v2f*)&xs[mlane * EMBED + k + kk];
    // B fragment from global W (row-major [K,N]); W is 64 KB -> L2-hot
    v2f b;
    b.x = W[(k + kk + 0) * EMBED + c0 + mlane];
    b.y = W[(k + kk + 1) * EMBED + c0 + mlane];
    acc = __builtin_amdgcn_wmma_f32_16x16x4_f32(false, a, false, b,
                                                (short)0, acc, false, false);
  }

  // Store D per the 16x16 f32 C/D layout.
  #pragma unroll
  for (int v = 0; v < 8; ++v) {
    h[(size_t)(r0 + v + hi * 8) * EMBED + (c0 + mlane)] = acc[v];
  }
}

// ---------------------------------------------------------------------------
// Zero agg (must run every call: launch is graph-replayed).
// ---------------------------------------------------------------------------
__global__ __launch_bounds__(256) void zero_f4(float4* __restrict__ p, int n4) {
  int i = blockIdx.x * blockDim.x + threadIdx.x;
  const int stride = gridDim.x * blockDim.x;
  const float4 z = {0.f, 0.f, 0.f, 0.f};
  for (; i < n4; i += stride) p[i] = z;
}

// ---------------------------------------------------------------------------
// One wave per edge: gather h[src] (512B contiguous b128 loads) and
// scatter-add into agg[dst] with non-returning f32 atomics (L2-resident).
// ---------------------------------------------------------------------------
__global__ __launch_bounds__(256) void edge_scatter(const float* __restrict__ h,
                                                    const int* __restrict__ esrc,
                                                    const int* __restrict__ edst,
                                                    float* __restrict__ agg,
                                                    int E) {
  const int lane = threadIdx.x & 31;
  int wid = (blockIdx.x * blockDim.x + threadIdx.x) >> 5;
  const int nw = (gridDim.x * blockDim.x) >> 5;
  for (int e = wid; e < E; e += nw) {
    const int s = __builtin_amdgcn_readfirstlane(esrc[e]);   // wave-uniform -> SGPR
    const int d = __builtin_amdgcn_readfirstlane(edst[e]);
    const float4 v = ((const float4*)(h + (size_t)s * EMBED))[lane];
    float* ap = agg + (size_t)d * EMBED + lane * 4;
    atomicAdd(ap + 0, v.x);   // global_atomic_add_f32, no return
    atomicAdd(ap + 1, v.y);
    atomicAdd(ap + 2, v.z);
    atomicAdd(ap + 3, v.w);
  }
}

// ---------------------------------------------------------------------------
// out = x + relu(agg + b), float4 streaming.
// ---------------------------------------------------------------------------
__global__ __launch_bounds__(256) void finalize(const float* __restrict__ x,
                                                const float* __restrict__ agg,
                                                const float* __restrict__ bias,
                                                float* __restrict__ out,
                                                int n4) {
  int i = blockIdx.x * blockDim.x + threadIdx.x;
  const int stride = gridDim.x * blockDim.x;
  for (; i < n4; i += stride) {
    const float4 xv = ((const float4*)x)[i];
    const float4 av = ((const float4*)agg)[i];
    const float4 bv = ((const float4*)bias)[i & (EMBED / 4 - 1)];
    float4 o;
    o.x = xv.x + fmaxf(av.x + bv.x, 0.f);
    o.y = xv.y + fmaxf(av.y + bv.y, 0.f);
    o.z = xv.z + fmaxf(av.z + bv.z, 0.f);
    o.w = xv.w + fmaxf(av.w + bv.w, 0.f);
    ((float4*)out)[i] = o;
  }
}

extern "C" void kernel_launch(void* const* d_in, const int* in_sizes, int n_in,
                              void* d_out, int out_size, void* d_ws, size_t ws_size,
                              hipStream_t stream) {
  const float* x    = (const float*)d_in[0];
  const int*   esrc = (const int*)d_in[1];
  const int*   edst = (const int*)d_in[2];
  const float* W    = (const float*)d_in[3];
  const float* bias = (const float*)d_in[4];
  float* out = (float*)d_out;

  const int N = in_sizes[0] / EMBED;   // 100000 nodes
  const int E = in_sizes[1];           // 1.6M edges

  float* h   = (float*)d_ws;                    // [N, 128]
  float* agg = h + (size_t)N * EMBED;           // [N, 128]

  const int n4 = (N * EMBED) / 4;

  zero_f4      <<<2048, 256, 0, stream>>>((float4*)agg, n4);
  gemm_xW_wmma <<<(N + 15) / 16, 256, 0, stream>>>(x, W, h, N);
  edge_scatter <<<4096, 256, 0, stream>>>(h, esrc, edst, agg, E);
  finalize     <<<2048, 256, 0, stream>>>(x, agg, bias, out, n4);
}